// HeteroRelConv_70763881169094
// MI455X (gfx1250) — compile-verified
//
#include <hip/hip_runtime.h>
#include <math.h>

typedef float v2f __attribute__((ext_vector_type(2)));
typedef float v8f __attribute__((ext_vector_type(8)));

#define NM 50000
#define NT 800000
#define NB 400000
#define NA 200000
#define NG 1000

// -------------------- scatter-add of neighbor features --------------------
// 16 threads per edge, each moving a float4 (row = 64 floats = 256B)
__global__ __launch_bounds__(256) void scatter_add_kernel(
    const float* __restrict__ xsrc, const int* __restrict__ ei, int n_e,
    float* __restrict__ agg, float* __restrict__ cnt)
{
    size_t total  = (size_t)n_e * 16;
    size_t stride = (size_t)gridDim.x * blockDim.x;
    for (size_t i = (size_t)blockIdx.x * blockDim.x + threadIdx.x; i < total; i += stride) {
        int e = (int)(i >> 4);
        int q = (int)(i & 15);
        int s = ei[e];
        int d = ei[(size_t)n_e + e];
        const float4 v = ((const float4*)(xsrc + (size_t)s * 64))[q];
        float* ap = agg + (size_t)d * 64 + q * 4;
        atomicAdd(ap + 0, v.x);
        atomicAdd(ap + 1, v.y);
        atomicAdd(ap + 2, v.z);
        atomicAdd(ap + 3, v.w);
        if (q == 0) atomicAdd(&cnt[d], 1.0f);
    }
}

// -------------------- WMMA helper: C[4] += (X*rscale) @ W^T tile ----------
// X: row-major [.,64] starting at the 16-row tile; lane holds row m.
// W: [64 out][64 in] row-major; B[k][n] = W[n][k].
__device__ __forceinline__ void gemm_acc(const float* __restrict__ Xrow, float rscale,
                                         const float* __restrict__ W,
                                         int nlo, int hi, v8f* c)
{
#pragma unroll
    for (int kb = 0; kb < 64; kb += 4) {
        const int kk = kb + 2 * hi;
        v2f a;
        a.x = Xrow[kk] * rscale;
        a.y = Xrow[kk + 1] * rscale;
#pragma unroll
        for (int nt = 0; nt < 4; ++nt) {
            const float* Wp = W + (size_t)((nt * 16 + nlo) << 6) + kk;
            v2f b;
            b.x = Wp[0];
            b.y = Wp[1];
            c[nt] = __builtin_amdgcn_wmma_f32_16x16x4_f32(
                false, a, false, b, (short)0, c[nt], false, false);
        }
    }
}

// -------------------- fused SAGE combine for one destination type ---------
// out = relu( sum_r( (agg_r/cnt_r) @ Wl_r^T + bl_r + xdst @ Wr_r^T ) )
__global__ __launch_bounds__(256) void sage_combine_kernel(
    const float* __restrict__ agg0, const float* __restrict__ cnt0,
    const float* __restrict__ Wl0, const float* __restrict__ bl0, const float* __restrict__ Wr0,
    const float* __restrict__ agg1, const float* __restrict__ cnt1,
    const float* __restrict__ Wl1, const float* __restrict__ bl1, const float* __restrict__ Wr1,
    const float* __restrict__ xdst, float* __restrict__ out,
    int n_tiles, int nrel)
{
    const int lane = threadIdx.x & 31;
    const int wave = threadIdx.x >> 5;
    const int tile = blockIdx.x * 8 + wave;
    if (tile >= n_tiles) return;              // wave-uniform exit (EXEC stays full)
    const int row_base = tile * 16;
    const int nlo = lane & 15;
    const int hi  = lane >> 4;
    const int m   = row_base + nlo;           // A-matrix row owned by this lane

    // accumulators: 4 n-tiles of 16 output features; init with bias(es)
    v8f c[4];
#pragma unroll
    for (int nt = 0; nt < 4; ++nt) {
        float bsum = bl0[nt * 16 + nlo];
        if (nrel > 1) bsum += bl1[nt * 16 + nlo];
#pragma unroll
        for (int v = 0; v < 8; ++v) c[nt][v] = bsum;
    }

    // relation 0: mean @ Wl0^T
    {
        float rinv = 1.0f / fmaxf(cnt0[m], 1.0f);
        gemm_acc(agg0 + (size_t)m * 64, rinv, Wl0, nlo, hi, c);
    }
    // relation 1: mean @ Wl1^T
    if (nrel > 1) {
        float rinv = 1.0f / fmaxf(cnt1[m], 1.0f);
        gemm_acc(agg1 + (size_t)m * 64, rinv, Wl1, nlo, hi, c);
    }
    // self terms: xdst @ Wr_r^T
    gemm_acc(xdst + (size_t)m * 64, 1.0f, Wr0, nlo, hi, c);
    if (nrel > 1) gemm_acc(xdst + (size_t)m * 64, 1.0f, Wr1, nlo, hi, c);

    // ReLU + store (C layout: vgpr v -> row v + 8*hi, lane -> col nlo of n-tile)
#pragma unroll
    for (int nt = 0; nt < 4; ++nt) {
#pragma unroll
        for (int v = 0; v < 8; ++v) {
            int r = row_base + v + 8 * hi;
            out[(size_t)r * 64 + nt * 16 + nlo] = fmaxf(c[nt][v], 0.0f);
        }
    }
}

// -------------------- pooling scatter (atoms -> graphs) -------------------
__global__ __launch_bounds__(256) void pool_scatter_kernel(
    const float* __restrict__ xa, const int* __restrict__ batch,
    float* __restrict__ pooled, float* __restrict__ gcnt, int n)
{
    size_t total  = (size_t)n * 16;
    size_t stride = (size_t)gridDim.x * blockDim.x;
    for (size_t i = (size_t)blockIdx.x * blockDim.x + threadIdx.x; i < total; i += stride) {
        int a = (int)(i >> 4);
        int q = (int)(i & 15);
        int g = batch[a];
        const float4 v = ((const float4*)(xa + (size_t)a * 64))[q];
        float* pp = pooled + (size_t)g * 64 + q * 4;
        atomicAdd(pp + 0, v.x);
        atomicAdd(pp + 1, v.y);
        atomicAdd(pp + 2, v.z);
        atomicAdd(pp + 3, v.w);
        if (q == 0) atomicAdd(&gcnt[g], 1.0f);
    }
}

// -------------------- MLP head: softplus(mean @ Pw^T + Pb) @ Ow^T + Ob ----
__global__ __launch_bounds__(64) void head_kernel(
    const float* __restrict__ pooled, const float* __restrict__ gcnt,
    const float* __restrict__ pw, const float* __restrict__ pb,
    const float* __restrict__ ow, const float* __restrict__ ob,
    float* __restrict__ out, int ngraphs)
{
    __shared__ float sh[64];
    int g = blockIdx.x;
    if (g >= ngraphs) return;
    int o = threadIdx.x;                       // 64 threads = one output feature each
    float rinv = 1.0f / fmaxf(gcnt[g], 1.0f);
    float s = pb[o];
    const float* pr = pooled + (size_t)g * 64;
    const float* wr = pw + (size_t)o * 64;
#pragma unroll 8
    for (int i = 0; i < 64; ++i) s += pr[i] * rinv * wr[i];
    // numerically stable softplus: max(x,0) + log1p(exp(-|x|))
    float hv = fmaxf(s, 0.0f) + log1pf(expf(-fabsf(s)));
    sh[o] = hv * ow[o];
    __syncthreads();
    if (o == 0) {
        float acc = ob[0];
        for (int i = 0; i < 64; ++i) acc += sh[i];
        out[g] = acc;
    }
}

// ==========================================================================
extern "C" void kernel_launch(void* const* d_in, const int* in_sizes, int n_in,
                              void* d_out, int out_size, void* d_ws, size_t ws_size,
                              hipStream_t stream) {
    const float* x_atom   = (const float*)d_in[0];
    const float* x_bond   = (const float*)d_in[1];
    const float* x_trip   = (const float*)d_in[2];
    const float* x_motif  = (const float*)d_in[3];
    const int*   ei_mm    = (const int*)d_in[4];
    const int*   ei_mt    = (const int*)d_in[5];
    const int*   ei_tt    = (const int*)d_in[6];
    const int*   ei_tb    = (const int*)d_in[7];
    const int*   ei_bb    = (const int*)d_in[8];
    const int*   ei_ba    = (const int*)d_in[9];
    const int*   ei_aa    = (const int*)d_in[10];
    const int*   batch    = (const int*)d_in[11];
    const float* Wl       = (const float*)d_in[12];   // [3,7,64,64]
    const float* bl       = (const float*)d_in[13];   // [3,7,64]
    const float* Wr       = (const float*)d_in[14];   // [3,7,64,64]
    const float* proj_w   = (const float*)d_in[15];
    const float* proj_b   = (const float*)d_in[16];
    const float* out_w    = (const float*)d_in[17];
    const float* out_b    = (const float*)d_in[18];
    float* out = (float*)d_out;

    const int E_mm = in_sizes[4] / 2,  E_mt = in_sizes[5] / 2, E_tt = in_sizes[6] / 2;
    const int E_tb = in_sizes[7] / 2,  E_bb = in_sizes[8] / 2, E_ba = in_sizes[9] / 2;
    const int E_aa = in_sizes[10] / 2;

    // ---- workspace layout (floats) ----
    float* base = (float*)d_ws;
    size_t off = 0;
    auto take = [&](size_t n) { float* r = base + off; off += n; return r; };
    float* Am = take((size_t)NM * 64);  float* At = take((size_t)NT * 64);
    float* Ab = take((size_t)NB * 64);  float* Aa = take((size_t)NA * 64);
    float* Bm = take((size_t)NM * 64);  float* Bt = take((size_t)NT * 64);
    float* Bb = take((size_t)NB * 64);  float* Ba = take((size_t)NA * 64);
    float* agg0 = take((size_t)NT * 64);
    float* agg1 = take((size_t)NT * 64);
    float* cnt0 = take((size_t)NT);
    float* cnt1 = take((size_t)NT);
    float* pooled = take((size_t)NG * 64);
    float* gcnt   = take((size_t)NG);
    (void)ws_size; (void)n_in; (void)out_size;

    auto scat = [&](const float* xsrc, const int* ei, int ne, float* agg, float* cnt, int ndst) {
        hipMemsetAsync(agg, 0, (size_t)ndst * 64 * sizeof(float), stream);
        hipMemsetAsync(cnt, 0, (size_t)ndst * sizeof(float), stream);
        size_t total = (size_t)ne * 16;
        int blocks = (int)((total + 255) / 256);
        scatter_add_kernel<<<blocks, 256, 0, stream>>>(xsrc, ei, ne, agg, cnt);
    };
    auto combine = [&](int l, int r0, int r1, const float* a0, const float* c0,
                       const float* a1, const float* c1,
                       const float* xdst, float* o, int n, int nrel) {
        const float* Wl0 = Wl + (size_t)(l * 7 + r0) * 4096;
        const float* bl0 = bl + (size_t)(l * 7 + r0) * 64;
        const float* Wr0 = Wr + (size_t)(l * 7 + r0) * 4096;
        const float* Wl1 = Wl + (size_t)(l * 7 + r1) * 4096;
        const float* bl1 = bl + (size_t)(l * 7 + r1) * 64;
        const float* Wr1 = Wr + (size_t)(l * 7 + r1) * 4096;
        int n_tiles = n / 16;
        int blocks = (n_tiles + 7) / 8;
        sage_combine_kernel<<<blocks, 256, 0, stream>>>(
            a0, c0, Wl0, bl0, Wr0, a1, c1, Wl1, bl1, Wr1, xdst, o, n_tiles, nrel);
    };

    const float* cm = x_motif; const float* ct = x_trip;
    const float* cb = x_bond;  const float* ca = x_atom;

    for (int l = 0; l < 3; ++l) {
        float* om = (l & 1) ? Bm : Am;  float* ot = (l & 1) ? Bt : At;
        float* ob2 = (l & 1) ? Bb : Ab; float* oa = (l & 1) ? Ba : Aa;

        // motif <- motif (rel 0)
        scat(cm, ei_mm, E_mm, agg0, cnt0, NM);
        combine(l, 0, 0, agg0, cnt0, agg0, cnt0, cm, om, NM, 1);

        // triplet <- motif (rel 1) + triplet (rel 2)
        scat(cm, ei_mt, E_mt, agg0, cnt0, NT);
        scat(ct, ei_tt, E_tt, agg1, cnt1, NT);
        combine(l, 1, 2, agg0, cnt0, agg1, cnt1, ct, ot, NT, 2);

        // bond <- triplet (rel 3) + bond (rel 4)
        scat(ct, ei_tb, E_tb, agg0, cnt0, NB);
        scat(cb, ei_bb, E_bb, agg1, cnt1, NB);
        combine(l, 3, 4, agg0, cnt0, agg1, cnt1, cb, ob2, NB, 2);

        // atom <- bond (rel 5) + atom (rel 6)
        scat(cb, ei_ba, E_ba, agg0, cnt0, NA);
        scat(ca, ei_aa, E_aa, agg1, cnt1, NA);
        combine(l, 5, 6, agg0, cnt0, agg1, cnt1, ca, oa, NA, 2);

        cm = om; ct = ot; cb = ob2; ca = oa;
    }

    // pooling (scatter-mean over graphs)
    hipMemsetAsync(pooled, 0, (size_t)NG * 64 * sizeof(float), stream);
    hipMemsetAsync(gcnt, 0, (size_t)NG * sizeof(float), stream);
    {
        size_t total = (size_t)NA * 16;
        int blocks = (int)((total + 255) / 256);
        pool_scatter_kernel<<<blocks, 256, 0, stream>>>(ca, batch, pooled, gcnt, NA);
    }
    head_kernel<<<NG, 64, 0, stream>>>(pooled, gcnt, proj_w, proj_b, out_w, out_b, out, NG);
}